// IdealScoreMachine_49460843381563
// MI455X (gfx1250) — compile-verified
//
#include <hip/hip_runtime.h>
#include <hip/hip_bf16.h>

typedef float v2f __attribute__((ext_vector_type(2)));
typedef float v8f __attribute__((ext_vector_type(8)));

#define B_ 64
#define D_ 3072
#define N_ 50000
#define NTILES (N_ / 16)     // 3125
#define DTILES (D_ / 16)     // 192
#define K1C 64               // K-chunk for GEMM1 (divides 3072)
#define K2C 80               // K-chunk for GEMM2 (divides 50000)

// ---------------------------------------------------------------------------
// Kernel 1: logw[64][50000] = (2*mu*xi - x2 - mu^2*i2) / (2*sigma^2),
//           xi = x @ images^T via fp32 WMMA; per-block max of logw.
// Grid: 3125 blocks (N tiles of 16), 128 threads (4 waves, M tiles of 16).
// ---------------------------------------------------------------------------
__global__ __launch_bounds__(128) void k1_logw(
    const float* __restrict__ x, const float* __restrict__ images,
    const float* __restrict__ mu_all, const float* __restrict__ sigma_all,
    const int* __restrict__ tptr,
    float* __restrict__ logw, float* __restrict__ blockmax)
{
    __shared__ float xs[B_][K1C + 1];   // x tile   (64 x 64), padded
    __shared__ float is[16][K1C + 1];   // img tile (16 x 64), padded
    __shared__ float x2s[B_];
    __shared__ float i2s[16];
    __shared__ float wmax[4];

    const int  t      = tptr[0];
    const float mu    = mu_all[t];
    const float sg    = sigma_all[t];
    const float inv2s2 = 1.0f / (2.0f * sg * sg);

    const int nbase = blockIdx.x * 16;
    const int tid   = threadIdx.x;
    const int lane  = tid & 31;
    const int wave  = tid >> 5;

    if (tid < B_) x2s[tid] = 0.0f;
    if (tid < 16) i2s[tid] = 0.0f;
    __syncthreads();

    float x2acc = 0.0f, i2acc = 0.0f;
    v8f acc = {};

    const int arow  = wave * 16 + (lane & 15);
    const int brow  = lane & 15;
    const int khalf = (lane >> 4) * 2;   // 0 for lanes 0-15, 2 for lanes 16-31

    for (int kc = 0; kc < D_; kc += K1C) {
        // Stage x chunk: 64x64 floats, 32 contiguous per thread (coalesced),
        // accumulate sum-of-squares for the row this thread touches.
        {
            const int row  = tid >> 1;
            const int col0 = (tid & 1) * 32;
            const float* src = x + (size_t)row * D_ + kc + col0;
            float s = 0.0f;
            #pragma unroll
            for (int j = 0; j < 32; ++j) {
                float v = src[j];
                xs[row][col0 + j] = v;
                s += v * v;
            }
            x2acc += s;
        }
        // Stage images chunk: 16x64 floats, 8 contiguous per thread.
        {
            const int row  = tid >> 3;
            const int col0 = (tid & 7) * 8;
            const float* src = images + (size_t)(nbase + row) * D_ + kc + col0;
            float s = 0.0f;
            #pragma unroll
            for (int j = 0; j < 8; ++j) {
                float v = src[j];
                is[row][col0 + j] = v;
                s += v * v;
            }
            i2acc += s;
        }
        __syncthreads();

        // fp32 WMMA: A = x tile rows (16x4), B = images^T (4x16).
        #pragma unroll
        for (int k = 0; k < K1C; k += 4) {
            v2f a, b;
            a.x = xs[arow][k + khalf + 0];
            a.y = xs[arow][k + khalf + 1];
            b.x = is[brow][k + khalf + 0];
            b.y = is[brow][k + khalf + 1];
            acc = __builtin_amdgcn_wmma_f32_16x16x4_f32(
                false, a, false, b, (short)0, acc, false, false);
        }
        __syncthreads();
    }

    atomicAdd(&x2s[tid >> 1], x2acc);
    atomicAdd(&i2s[tid >> 3], i2acc);
    __syncthreads();

    // Epilogue: C element (vgpr r, lane l) -> M = wave*16 + r + 8*(l>>4),
    // N = nbase + (l&15).
    const float i2v   = i2s[lane & 15];
    const float mi2   = mu * mu * i2v;
    const int   mrow0 = wave * 16 + ((lane >> 4) ? 8 : 0);
    const int   ncol  = nbase + (lane & 15);
    float lmax = -3.402823466e+38f;
    #pragma unroll
    for (int r = 0; r < 8; ++r) {
        const int m = mrow0 + r;
        float lw = (2.0f * mu * acc[r] - x2s[m] - mi2) * inv2s2;
        logw[(size_t)m * N_ + ncol] = lw;
        lmax = fmaxf(lmax, lw);
    }
    #pragma unroll
    for (int off = 16; off > 0; off >>= 1)
        lmax = fmaxf(lmax, __shfl_xor(lmax, off, 32));
    if (lane == 0) wmax[wave] = lmax;
    __syncthreads();
    if (tid == 0)
        blockmax[blockIdx.x] =
            fmaxf(fmaxf(wmax[0], wmax[1]), fmaxf(wmax[2], wmax[3]));
}

// ---------------------------------------------------------------------------
// Kernel 2: global max of blockmax[3125] -> gmax[0]
// ---------------------------------------------------------------------------
__global__ __launch_bounds__(256) void k_maxreduce(
    const float* __restrict__ blockmax, int n, float* __restrict__ gmax)
{
    __shared__ float sm[256];
    float m = -3.402823466e+38f;
    for (int i = threadIdx.x; i < n; i += 256) m = fmaxf(m, blockmax[i]);
    sm[threadIdx.x] = m;
    __syncthreads();
    for (int s = 128; s > 0; s >>= 1) {
        if (threadIdx.x < s)
            sm[threadIdx.x] = fmaxf(sm[threadIdx.x], sm[threadIdx.x + s]);
        __syncthreads();
    }
    if (threadIdx.x == 0) gmax[0] = sm[0];
}

// ---------------------------------------------------------------------------
// Kernel 3: S = w @ images with w = exp(logw - m) computed during staging;
// sum_w accumulated on the fly; fused epilogue -> score.
// Grid: 192 blocks (D tiles of 16), 128 threads (4 waves, M tiles of 16).
// ---------------------------------------------------------------------------
__global__ __launch_bounds__(128) void k2_score(
    const float* __restrict__ x, const float* __restrict__ images,
    const float* __restrict__ mu_all, const float* __restrict__ sigma_all,
    const int* __restrict__ tptr,
    const float* __restrict__ logw, const float* __restrict__ gmax,
    float* __restrict__ out)
{
    __shared__ float wt[B_][K2C + 1];   // w tile (64 x 80), padded
    __shared__ float bt[K2C][16 + 1];   // images chunk [k][n], padded
    __shared__ float sws[B_];

    const int   t   = tptr[0];
    const float mu  = mu_all[t];
    const float sg  = sigma_all[t];
    const float gm  = gmax[0];

    const int dbase = blockIdx.x * 16;
    const int tid   = threadIdx.x;
    const int lane  = tid & 31;
    const int wave  = tid >> 5;

    if (tid < B_) sws[tid] = 0.0f;
    __syncthreads();

    float swacc = 0.0f;
    v8f acc = {};

    const int arow  = wave * 16 + (lane & 15);
    const int ncol  = lane & 15;
    const int khalf = (lane >> 4) * 2;

    for (int kc = 0; kc < N_; kc += K2C) {
        // Stage w chunk: 64x80, 40 contiguous per thread; exp during staging.
        {
            const int row  = tid >> 1;
            const int col0 = (tid & 1) * 40;
            const float* src = logw + (size_t)row * N_ + kc + col0;
            float s = 0.0f;
            #pragma unroll
            for (int j = 0; j < 40; ++j) {
                float v = __expf(src[j] - gm);
                wt[row][col0 + j] = v;
                s += v;
            }
            swacc += s;
        }
        // Stage images chunk: rows kc..kc+79, 16 cols each (contiguous).
        if (tid < K2C) {
            const float* src = images + (size_t)(kc + tid) * D_ + dbase;
            #pragma unroll
            for (int j = 0; j < 16; ++j) bt[tid][j] = src[j];
        }
        __syncthreads();

        #pragma unroll
        for (int k = 0; k < K2C; k += 4) {
            v2f a, b;
            a.x = wt[arow][k + khalf + 0];
            a.y = wt[arow][k + khalf + 1];
            b.x = bt[k + khalf + 0][ncol];
            b.y = bt[k + khalf + 1][ncol];
            acc = __builtin_amdgcn_wmma_f32_16x16x4_f32(
                false, a, false, b, (short)0, acc, false, false);
        }
        __syncthreads();
    }

    atomicAdd(&sws[tid >> 1], swacc);
    __syncthreads();

    const float inv_s2 = 1.0f / (sg * sg);
    const int   mrow0  = wave * 16 + ((lane >> 4) ? 8 : 0);
    const int   d      = dbase + (lane & 15);
    #pragma unroll
    for (int r = 0; r < 8; ++r) {
        const int m = mrow0 + r;
        const float sw = sws[m];
        const float xv = x[(size_t)m * D_ + d];
        out[(size_t)m * D_ + d] = -(xv - mu * acc[r] / sw) * inv_s2;
    }
}

// ---------------------------------------------------------------------------
extern "C" void kernel_launch(void* const* d_in, const int* in_sizes, int n_in,
                              void* d_out, int out_size, void* d_ws, size_t ws_size,
                              hipStream_t stream) {
    const float* x         = (const float*)d_in[0];
    const float* images    = (const float*)d_in[1];
    const float* mu_all    = (const float*)d_in[2];
    const float* sigma_all = (const float*)d_in[3];
    const int*   t         = (const int*)d_in[4];
    float* out = (float*)d_out;

    float* logw     = (float*)d_ws;                       // 64*50000 floats = 12.8 MB
    float* blockmax = logw + (size_t)B_ * N_;             // 3125 floats
    float* gmax     = blockmax + 3200;                    // 1 float

    k1_logw<<<NTILES, 128, 0, stream>>>(x, images, mu_all, sigma_all, t,
                                        logw, blockmax);
    k_maxreduce<<<1, 256, 0, stream>>>(blockmax, NTILES, gmax);
    k2_score<<<DTILES, 128, 0, stream>>>(x, images, mu_all, sigma_all, t,
                                         logw, gmax, out);
}